// AddShift_mp_linear_module_60035052863994
// MI455X (gfx1250) — compile-verified
//
#include <hip/hip_runtime.h>
#include <hip/hip_bf16.h>

// ---------------------------------------------------------------------------
// Problem constants (from the reference):
//   BIG_K=51, SMALL_K=5  -> NK=11, C_OUT=96, C_IN=1056, GROUP_IN=4
//   B=8, HOUT=WOUT=56, EXTRA_PAD=2 -> HIN=WIN=60
// Memory-bound gather kernel: stage 11 contiguous 60x60 f32 planes (158.4 KB)
// per (b, c_out) block into LDS via CDNA5 async global->LDS DMA, then do all
// 92 taps/pixel from LDS.  x is read from HBM exactly once (~122 MB), output
// 29 MB -> roofline ~6.5 us @ 23.3 TB/s.
// ---------------------------------------------------------------------------

#define KC_OUT   96
#define KNK      11
#define KG       4
#define KC_IN    (KC_OUT * KNK)     // 1056
#define KB       8
#define KHOUT    56
#define KWOUT    56
#define KHIN     60
#define KWIN     60
#define KEP      2
#define KPLANE   (KHIN * KWIN)      // 3600 floats
#define KLDSF    (KNK * KPLANE)     // 39600 floats = 158400 bytes
#define KSMEMB   (KLDSF * 4)
#define KNTHREADS 448               // 14 waves (wave32); 3136 px = 7 px/thread
#define KCHUNKS  (KLDSF / 4)        // 9900 float4 chunks = 448*22 + 44
#define KNOUT    ((size_t)KB * KC_OUT * KHOUT * KWOUT)  // 2,408,448 per output

// --- gfx1250 async global->LDS path (guarded; fall back to plain copy) ------
#if defined(__has_builtin)
#  if __has_builtin(__builtin_amdgcn_global_load_async_to_lds_b128)
#    define USE_ASYNC_LDS 1
#  else
#    define USE_ASYNC_LDS 0
#  endif
#  if __has_builtin(__builtin_amdgcn_s_wait_asynccnt)
#    define HAVE_WAIT_ASYNC 1
#  else
#    define HAVE_WAIT_ASYNC 0
#  endif
#else
#  define USE_ASYNC_LDS 0
#  define HAVE_WAIT_ASYNC 0
#endif

// Builtin signature (from hipcc diagnostic): (global v4i*, lds v4i*, imm, imm)
typedef int v4i_ __attribute__((__vector_size__(16)));
typedef __attribute__((address_space(1))) v4i_* gas_v4i_p;
typedef __attribute__((address_space(3))) v4i_* las_v4i_p;

__device__ __forceinline__ void copy16_g2l(const float* g, float* l) {
#if USE_ASYNC_LDS
    __builtin_amdgcn_global_load_async_to_lds_b128((gas_v4i_p)g, (las_v4i_p)l, 0, 0);
#else
    *reinterpret_cast<float4*>(l) = *reinterpret_cast<const float4*>(g);
#endif
}

__global__ __launch_bounds__(KNTHREADS)
void addshift_fused_kernel(const float* __restrict__ x,
                           const float* __restrict__ w1,
                           const float* __restrict__ w2,
                           const float* __restrict__ w3,
                           const int*   __restrict__ pad_hv,
                           const int*   __restrict__ idx_id,
                           float*       __restrict__ out)
{
    extern __shared__ float sm[];                 // KNK planes of 60x60 f32
    __shared__ int4  prm[KNK * KG];               // {w1 bits, w2 bits, shiftW, shiftH}
    __shared__ float w3s[KG];
    __shared__ int   idks[KG];

    const int tid = threadIdx.x;
    const int co  = blockIdx.x;                   // 0..95
    const int bi  = blockIdx.y;                   // 0..7

    // ---- kick off async DMA of 11 contiguous planes: global -> LDS ---------
    // Issued FIRST so the 158.4 KB transfer overlaps the parameter loads.
    const float* __restrict__ src =
        x + (size_t)(bi * KC_IN + co * KNK) * KPLANE;   // 16B aligned
#pragma unroll
    for (int r = 0; r < 22; ++r) {
        const int i = tid + r * KNTHREADS;
        copy16_g2l(src + 4 * i, sm + 4 * i);
    }
    if (tid < KCHUNKS - 22 * KNTHREADS) {               // tail: 44 chunks
        const int i = tid + 22 * KNTHREADS;
        copy16_g2l(src + 4 * i, sm + 4 * i);
    }

    // ---- per-block tap parameters into LDS (overlaps the DMA) --------------
    if (tid < KNK * KG) {
        const int k = tid >> 2, g = tid & 3;
        const int c = co * KNK + k;
        int4 pr;
        pr.x = __float_as_int(w1[g * KC_IN + c]);
        pr.y = __float_as_int(w2[g * KC_IN + c]);
        pr.z = KEP + pad_hv[c * (2 * KG) + g];        // horizontal shift
        pr.w = KEP + pad_hv[c * (2 * KG) + KG + g];   // vertical shift
        prm[tid] = pr;
    }
    if (tid < KG) {
        w3s[tid]  = w3[tid * KC_OUT + co];
        idks[tid] = idx_id[co * KG + tid] - co * KNK; // local k in [0,11)
    }

#if USE_ASYNC_LDS
#  if HAVE_WAIT_ASYNC
    __builtin_amdgcn_s_wait_asynccnt(0);
#  else
    asm volatile("s_wait_asynccnt 0" ::: "memory");
#  endif
#endif
    __syncthreads();

    // ---- compute: each thread owns column w, rows h0+8j (j=0..6) -----------
    const int w  = tid % KWOUT;          // 0..55
    const int h0 = tid / KWOUT;          // 0..7
    const int col_c = w + KEP;           // column for vertical + identity taps

    float accH[7], accV[7], accI[7];
#pragma unroll
    for (int j = 0; j < 7; ++j) { accH[j] = 0.f; accV[j] = 0.f; accI[j] = 0.f; }

    for (int k = 0; k < KNK; ++k) {
        const float* pl = sm + k * KPLANE;
#pragma unroll
        for (int g = 0; g < KG; ++g) {
            const int4 pr = prm[k * KG + g];
            // horizontal tap: fixed per thread (w fixed), reused for all 7 rows
            const int   iw  = w + pr.z;
            const float w1m = ((unsigned)iw < (unsigned)KWIN)
                                ? __int_as_float(pr.x) : 0.0f;
            const int   iwc = min(max(iw, 0), KWIN - 1);
            const float w2v = __int_as_float(pr.y);
#pragma unroll
            for (int j = 0; j < 7; ++j) {
                const int h = h0 + 8 * j;
                accH[j] += w1m * pl[(h + KEP) * KWIN + iwc];
                const int   ih  = h + pr.w;
                const float w2m = ((unsigned)ih < (unsigned)KHIN) ? w2v : 0.0f;
                const int   ihc = min(max(ih, 0), KHIN - 1);
                accV[j] += w2m * pl[ihc * KWIN + col_c];
            }
        }
    }

    // identity path: 4 taps from channels inside this co block
#pragma unroll
    for (int g = 0; g < KG; ++g) {
        const float* pl = sm + idks[g] * KPLANE;
        const float  wv = w3s[g];
#pragma unroll
        for (int j = 0; j < 7; ++j)
            accI[j] += wv * pl[(h0 + 8 * j + KEP) * KWIN + col_c];
    }

    // ---- store (out_h, out_v, out_i concatenated flat) ---------------------
    const size_t base = ((size_t)(bi * KC_OUT + co) * KHOUT) * KWOUT;
#pragma unroll
    for (int j = 0; j < 7; ++j) {
        const size_t o = base + (size_t)(h0 + 8 * j) * KWOUT + w;
        out[o]             = accH[j];
        out[KNOUT + o]     = accV[j];
        out[2 * KNOUT + o] = accI[j];
    }
}

extern "C" void kernel_launch(void* const* d_in, const int* in_sizes, int n_in,
                              void* d_out, int out_size, void* d_ws, size_t ws_size,
                              hipStream_t stream) {
    (void)in_sizes; (void)n_in; (void)out_size; (void)d_ws; (void)ws_size;
    const float* x      = (const float*)d_in[0];
    const float* w1     = (const float*)d_in[1];
    const float* w2     = (const float*)d_in[2];
    const float* w3     = (const float*)d_in[3];
    const int*   pad_hv = (const int*)d_in[4];
    const int*   idx_id = (const int*)d_in[5];
    float*       out    = (float*)d_out;

    // 158.4 KB dynamic LDS per block (>64 KB default cap) — opt in every call.
    (void)hipFuncSetAttribute(reinterpret_cast<const void*>(addshift_fused_kernel),
                              hipFuncAttributeMaxDynamicSharedMemorySize, KSMEMB);

    dim3 grid(KC_OUT, KB);   // (co, b) = 96 x 8 = 768 blocks
    addshift_fused_kernel<<<grid, KNTHREADS, KSMEMB, stream>>>(
        x, w1, w2, w3, pad_hv, idx_id, out);
}